// GraphNetLayers_79293686218961
// MI455X (gfx1250) — compile-verified
//
#include <hip/hip_runtime.h>

// Problem constants (match reference)
#define NN 10000      // nodes
#define NE 320000     // edges
#define CD 256        // feature dim
#define NL 3          // layers
#define DD 20         // KAF dictionary size
// KAF constants: interval = 2*4/(D-1), gamma = 0.5/(2*interval)^2
#define KAF_INTERVAL (8.0f / 19.0f)
#define KAF_GAMMA    (0.5f / ((2.0f * KAF_INTERVAL) * (2.0f * KAF_INTERVAL)))

typedef float v2f __attribute__((ext_vector_type(2)));
typedef float v8f __attribute__((ext_vector_type(8)));

// ---------------------------------------------------------------------------
// Zero-fill (d_ws is poisoned 0xAA by harness; we must init our accumulators)
// ---------------------------------------------------------------------------
__global__ void zero_kernel(float* __restrict__ p, int n) {
  int i = blockIdx.x * blockDim.x + threadIdx.x;
  if (i < n) p[i] = 0.0f;
}

// ---------------------------------------------------------------------------
// Per-destination degree count + edge_attr aggregation (graph-constant, done
// once per launch). cnt[dst] += 1 ; A[dst,0:2] += edge_attr[e,0:2]
// ---------------------------------------------------------------------------
__global__ void edge_deg_kernel(const int* __restrict__ ei,
                                const float* __restrict__ ea,
                                float* __restrict__ cnt,
                                float* __restrict__ Aagg) {
  int e = blockIdx.x * blockDim.x + threadIdx.x;
  if (e >= NE) return;
  int dst = ei[NE + e];           // edge_index[1, e]
  atomicAdd(&cnt[dst], 1.0f);
  atomicAdd(&Aagg[dst * 2 + 0], ea[e * 2 + 0]);
  atomicAdd(&Aagg[dst * 2 + 1], ea[e * 2 + 1]);
}

__global__ void inv_kernel(const float* __restrict__ cnt, float* __restrict__ inv) {
  int i = blockIdx.x * blockDim.x + threadIdx.x;
  if (i < NN) {
    float c = cnt[i];
    inv[i] = (c > 0.0f) ? (1.0f / c) : 0.0f;
  }
}

// ---------------------------------------------------------------------------
// Edge scatter: S[dst,:] += h[src,:]. One thread per (edge, 4-channel chunk):
// float4 gather from h (L2-resident, 10.2MB) + 4 f32 atomic adds into S.
// ---------------------------------------------------------------------------
__global__ void scatter_kernel(const int* __restrict__ ei,
                               const float* __restrict__ h,
                               float* __restrict__ S) {
  int t = blockIdx.x * blockDim.x + threadIdx.x;   // < NE * (CD/4) = 20.48M
  if (t >= NE * (CD / 4)) return;
  int e = t >> 6;               // edge index      (CD/4 == 64 chunks)
  int c = (t & 63) << 2;        // channel base
  int src = ei[e];              // edge_index[0, e]
  int dst = ei[NE + e];         // edge_index[1, e]
  float4 v = *reinterpret_cast<const float4*>(h + (size_t)src * CD + c);
  float* d = S + (size_t)dst * CD + c;
  atomicAdd(d + 0, v.x);
  atomicAdd(d + 1, v.y);
  atomicAdd(d + 2, v.z);
  atomicAdd(d + 3, v.w);
}

// ---------------------------------------------------------------------------
// Fused layer kernel:
//   pre[m,n] = inv[m]*(S @ Wn)[m,n] + (H @ Wx)[m,n]
//            + inv[m]*(A[m,0]*We[0,n] + A[m,1]*We[1,n]) + bias[n]
//   out[m,n] = sum_d alpha[n,d] * exp(-gamma*(pre - dict[d])^2)
//
// One wave32 per 16x16 output tile, V_WMMA_F32_16X16X4_F32, K=256 per GEMM.
// A-fragment (16x4 f32, 2 VGPR): lane L -> row m=L&15, k = kk + 2*(L>>4)+v
// B-fragment (4x16 f32, 2 VGPR): lane L -> col n=L&15, same k mapping
// C/D (16x16 f32, 8 VGPR):       lane L -> col n=L&15, rows r + 8*(L>>4)
// inv_cnt row scaling is folded into the A-fragment loads of the S GEMM.
// ---------------------------------------------------------------------------
__global__ __launch_bounds__(256)
void gemm_kaf_kernel(const float* __restrict__ S,
                     const float* __restrict__ H,
                     const float* __restrict__ inv,
                     const float* __restrict__ Aagg,
                     const float* __restrict__ Wn,    // neigh_w[l]  [CD,CD]
                     const float* __restrict__ Wx,    // node_w[l]   [CD,CD]
                     const float* __restrict__ We,    // edge_w[l]   [2,CD]
                     const float* __restrict__ bias,  // [CD]
                     const float* __restrict__ alpha, // [CD,DD]
                     float* __restrict__ out) {
  const int lane = threadIdx.x & 31;
  const int wave = threadIdx.x >> 5;
  const int tile = blockIdx.x * 8 + wave;          // 10000/16 * 256/16 = 10000 tiles
  const int tilesN = CD / 16;                      // 16 column tiles
  const int tm = (tile / tilesN) * 16;             // row base   (N%16==0, no tail)
  const int tn = (tile % tilesN) * 16;             // col base
  const int ln = lane & 15;
  const int lh = lane >> 4;                        // 0 or 1

  const int n  = tn + ln;                          // this lane's output column
  const int am = tm + ln;                          // this lane's A-fragment row
  const float inv_am = inv[am];

  // Accumulator init: rank-2 edge_attr update (inv-scaled) + bias.
  const float we0 = We[n];
  const float we1 = We[CD + n];
  const float bn  = bias[n];
  v8f acc;
#pragma unroll
  for (int r = 0; r < 8; ++r) {
    const int m = tm + r + 8 * lh;
    acc[r] = inv[m] * (Aagg[m * 2 + 0] * we0 + Aagg[m * 2 + 1] * we1) + bn;
  }

  // GEMM 1: (inv .* S) @ neigh_w   — scaling folded into A-fragment load.
#pragma unroll 4
  for (int k = 0; k < CD; k += 4) {
    const int ka = k + 2 * lh;
    v2f a, b;
    a[0] = S[(size_t)am * CD + ka]     * inv_am;
    a[1] = S[(size_t)am * CD + ka + 1] * inv_am;
    b[0] = Wn[(size_t)ka * CD + n];
    b[1] = Wn[(size_t)(ka + 1) * CD + n];
    acc = __builtin_amdgcn_wmma_f32_16x16x4_f32(false, a, false, b,
                                                (short)0, acc, false, false);
  }

  // GEMM 2: H @ node_w
#pragma unroll 4
  for (int k = 0; k < CD; k += 4) {
    const int ka = k + 2 * lh;
    v2f a, b;
    a[0] = H[(size_t)am * CD + ka];
    a[1] = H[(size_t)am * CD + ka + 1];
    b[0] = Wx[(size_t)ka * CD + n];
    b[1] = Wx[(size_t)(ka + 1) * CD + n];
    acc = __builtin_amdgcn_wmma_f32_16x16x4_f32(false, a, false, b,
                                                (short)0, acc, false, false);
  }

  // KAF epilogue: per-channel Gaussian mixture (alpha row is per-column n).
  float al[DD];
#pragma unroll
  for (int d = 0; d < DD; ++d) al[d] = alpha[n * DD + d];

#pragma unroll
  for (int r = 0; r < 8; ++r) {
    const int m = tm + r + 8 * lh;
    const float s = acc[r];
    float o = 0.0f;
#pragma unroll
    for (int d = 0; d < DD; ++d) {
      const float dp = -4.0f + (float)d * KAF_INTERVAL;
      const float t  = s - dp;
      o += al[d] * __expf(-KAF_GAMMA * t * t);
    }
    out[(size_t)m * CD + n] = o;
  }
}

// ---------------------------------------------------------------------------
// Launch: zero/deg/inv once, then L x { zero S, scatter, fused GEMM+KAF }.
// h chain: x -> d_out -> h_mid -> d_out  (final result lands in d_out).
// ---------------------------------------------------------------------------
extern "C" void kernel_launch(void* const* d_in, const int* in_sizes, int n_in,
                              void* d_out, int out_size, void* d_ws, size_t ws_size,
                              hipStream_t stream) {
  const float* x       = (const float*)d_in[0];
  const int*   ei      = (const int*)  d_in[1];
  const float* ea      = (const float*)d_in[2];
  const float* node_w  = (const float*)d_in[3];
  const float* edge_w  = (const float*)d_in[4];
  const float* neigh_w = (const float*)d_in[5];
  const float* bias    = (const float*)d_in[6];
  const float* alpha   = (const float*)d_in[7];
  float* outp = (float*)d_out;

  // Workspace layout (all offsets 16B-aligned): ~20.6 MB total.
  float* h_mid = (float*)d_ws;               // NN*CD
  float* S     = h_mid + (size_t)NN * CD;    // NN*CD
  float* Aagg  = S     + (size_t)NN * CD;    // NN*2
  float* cnt   = Aagg  + (size_t)NN * 2;     // NN
  float* invp  = cnt   + NN;                 // NN

  const int B = 256;

  // Graph-constant preprocessing (recomputed every call: deterministic).
  zero_kernel<<<(NN + B - 1) / B, B, 0, stream>>>(cnt, NN);
  zero_kernel<<<(2 * NN + B - 1) / B, B, 0, stream>>>(Aagg, 2 * NN);
  edge_deg_kernel<<<(NE + B - 1) / B, B, 0, stream>>>(ei, ea, cnt, Aagg);
  inv_kernel<<<(NN + B - 1) / B, B, 0, stream>>>(cnt, invp);

  const float* h_cur[NL]  = { x,    outp,  h_mid };
  float*       h_next[NL] = { outp, h_mid, outp  };

  const int scatterThreads = NE * (CD / 4);
  const int gemmBlocks = (NN / 16) * (CD / 16) / 8;   // 10000 tiles / 8 waves

  for (int l = 0; l < NL; ++l) {
    zero_kernel<<<(NN * CD + B - 1) / B, B, 0, stream>>>(S, NN * CD);
    scatter_kernel<<<(scatterThreads + B - 1) / B, B, 0, stream>>>(ei, h_cur[l], S);
    gemm_kaf_kernel<<<gemmBlocks, B, 0, stream>>>(
        S, h_cur[l], invp, Aagg,
        neigh_w + (size_t)l * CD * CD,
        node_w  + (size_t)l * CD * CD,
        edge_w  + (size_t)l * 2 * CD,
        bias    + (size_t)l * CD,
        alpha   + (size_t)l * CD * DD,
        h_next[l]);
  }
}